// GraphNet_34660386079059
// MI455X (gfx1250) — compile-verified
//
#include <hip/hip_runtime.h>

#define BB 8
#define NN 128
#define DD 256
#define HH 10
#define KIN 64

typedef __attribute__((ext_vector_type(2))) float v2f;
typedef __attribute__((ext_vector_type(8))) float v8f;

__device__ __forceinline__ float leaky(float x) { return x >= 0.f ? x : 0.01f * x; }

__device__ __forceinline__ v8f wmma_f32(v2f a, v2f b, v8f c) {
  // V_WMMA_F32_16X16X4_F32: D = A(16x4) * B(4x16) + C(16x16), full f32
  return __builtin_amdgcn_wmma_f32_16x16x4_f32(false, a, false, b, (short)0, c, false, false);
}

// One block per (b,i). 8 waves. Computes h_out[b,i,0:256].
// LAYER1: feat tile built by WMMA GEMM feature[b,i,:,:] (128x64) @ l1_w (64x256) + bias.
// else:   feat tile = leaky(P[b,j,:] + Q[b,i,:] + bias).
template <bool LAYER1>
__global__ __launch_bounds__(256) void gat_kernel(
    const float* __restrict__ Pm, const float* __restrict__ Qm,
    const float* __restrict__ feature, const float* __restrict__ l1w,
    const float* __restrict__ bias, const float* __restrict__ valid,
    const float* __restrict__ mw, const float* __restrict__ mb,
    float* __restrict__ hout) {
  __shared__ float featT[16][DD + 1];   // 16 j-rows x 256 d (padded)
  __shared__ float paw[8][16][16];      // per-wave partial attention scores
  __shared__ float wt[16][17];          // masked exp weights for current j-tile
  __shared__ float den[16];             // per-head softmax denominator

  const int tid = threadIdx.x;
  const int wave = tid >> 5;
  const int lane = tid & 31;
  const int halfsel = lane >> 4;   // 0: lanes 0-15, 1: lanes 16-31
  const int lrow = lane & 15;

  const int b = blockIdx.x / NN;
  const int i = blockIdx.x % NN;

  if (tid < 16) den[tid] = 0.f;

  float qv = 0.f, lbv = 0.f;
  if constexpr (!LAYER1) {
    qv = Qm[(size_t)(b * NN + i) * DD + tid];
    lbv = bias[tid];
  }

  v8f zero = {};
  v8f chn[2];   // persistent numerator accumulators: wave owns d-cols [wave*32, wave*32+32)
  chn[0] = zero;
  chn[1] = zero;

  __syncthreads();

  for (int jt = 0; jt < 8; ++jt) {
    // ---- phase 1: build feat tile (16 x 256) in LDS ----
    if constexpr (!LAYER1) {
      #pragma unroll 4
      for (int jj = 0; jj < 16; ++jj) {
        int j = jt * 16 + jj;
        float p = Pm[(size_t)(b * NN + j) * DD + tid];
        featT[jj][tid] = leaky(p + qv + lbv);
      }
    } else {
      const float* arow =
          feature + ((size_t)(b * NN + i) * NN + (jt * 16 + lrow)) * KIN + halfsel * 2;
      #pragma unroll
      for (int t = 0; t < 2; ++t) {
        int d0 = wave * 32 + t * 16;
        int n = d0 + lrow;
        v8f acc = zero;
        #pragma unroll
        for (int k = 0; k < KIN; k += 4) {
          v2f a, bm;
          a.x = arow[k];
          a.y = arow[k + 1];
          const float* bp = l1w + (size_t)(k + halfsel * 2) * DD + n;
          bm.x = bp[0];
          bm.y = bp[DD];
          acc = wmma_f32(a, bm, acc);
        }
        float bv = bias[n];
        #pragma unroll
        for (int v = 0; v < 8; ++v) {
          int m = v + halfsel * 8;
          featT[m][n] = leaky(acc[v] + bv);
        }
      }
    }
    __syncthreads();

    // ---- phase 2: attention scores aw[16j x 16h] = featT @ mw^T, K split over waves ----
    {
      v8f acc = zero;
      const int kb = wave * 32;
      const bool hv = lrow < HH;
      #pragma unroll
      for (int kk = 0; kk < 32; kk += 4) {
        int k = kb + kk + halfsel * 2;
        v2f a, bm;
        a.x = featT[lrow][k];
        a.y = featT[lrow][k + 1];
        bm.x = hv ? mw[lrow * DD + k] : 0.f;
        bm.y = hv ? mw[lrow * DD + k + 1] : 0.f;
        acc = wmma_f32(a, bm, acc);
      }
      #pragma unroll
      for (int v = 0; v < 8; ++v) paw[wave][v + halfsel * 8][lrow] = acc[v];
    }
    __syncthreads();

    // reduce K-partials, apply leaky/exp/mask -> wt[jj][h]
    {
      int m = tid >> 4, n = tid & 15;
      float s = 0.f;
      #pragma unroll
      for (int wv = 0; wv < 8; ++wv) s += paw[wv][m][n];
      float wval = 0.f;
      if (n < HH) {
        float awv = s + mb[n];
        wval = __expf(leaky(awv)) * valid[(size_t)(b * NN + i) * NN + (jt * 16 + m)];
      }
      wt[m][n] = wval;
    }
    __syncthreads();

    if (tid < 16) {
      float s = 0.f;
      for (int m = 0; m < 16; ++m) s += wt[m][tid];
      den[tid] += s;
    }

    // ---- phase 4: numerator accumulation hnum[16h x 256d] += wt^T @ featT ----
    #pragma unroll
    for (int t = 0; t < 2; ++t) {
      int d0 = wave * 32 + t * 16;
      int n = d0 + lrow;
      v8f acc = chn[t];
      #pragma unroll
      for (int k = 0; k < 16; k += 4) {
        int kk = k + halfsel * 2;
        v2f a, bm;
        a.x = wt[kk][lrow];       // A[m=h][k=jj] = wt[jj][h]
        a.y = wt[kk + 1][lrow];
        bm.x = featT[kk][n];      // B[k=jj][n=d]
        bm.y = featT[kk + 1][n];
        acc = wmma_f32(a, bm, acc);
      }
      chn[t] = acc;
    }
    __syncthreads();
  }

  // ---- finalize: normalize per head, max over real heads, write h_out ----
  __syncthreads();
  #pragma unroll
  for (int t = 0; t < 2; ++t) {
    int d0 = wave * 32 + t * 16;
    int n = d0 + lrow;
    float mx = -3.4e38f;
    #pragma unroll
    for (int v = 0; v < 8; ++v) {
      int hh = v + halfsel * 8;   // C layout: vgpr v holds head v (+8 for upper lanes)
      if (hh < HH) mx = fmaxf(mx, chn[t][v] / (den[hh] + 1e-9f));
    }
    mx = fmaxf(mx, __shfl_xor(mx, 16, 32));   // combine heads 0-7 with 8-9
    if (halfsel == 0) hout[(size_t)(b * NN + i) * DD + n] = mx;
  }
}

// P = h @ lw[0:256,:]   (j-dependent half of pairwise_cat)
// Q = h @ lw[256:512,:] (i-dependent half)
// grid: 64 row-tiles (16 rows) x 4 col-groups (128 of 512 out-cols); 8 waves -> 1 tile each.
__global__ __launch_bounds__(256) void pq_kernel(const float* __restrict__ h,
                                                 const float* __restrict__ lw,
                                                 float* __restrict__ P,
                                                 float* __restrict__ Q) {
  __shared__ float At[16][DD + 1];
  const int tid = threadIdx.x;
  const int wave = tid >> 5;
  const int lane = tid & 31;
  const int halfsel = lane >> 4;
  const int lrow = lane & 15;
  const int rt = blockIdx.x >> 2;
  const int cg = blockIdx.x & 3;

  #pragma unroll 4
  for (int m = 0; m < 16; ++m)
    At[m][tid] = h[(size_t)(rt * 16 + m) * DD + tid];
  __syncthreads();

  int nt = cg * 128 + wave * 16;            // 0..496 over combined 512 out-cols
  bool isQ = nt >= DD;
  const float* wb = lw + (isQ ? (size_t)DD * DD : 0);
  int ncol = (isQ ? nt - DD : nt) + lrow;
  v8f acc = {};
  #pragma unroll 8
  for (int k = 0; k < DD; k += 4) {
    int kk = k + halfsel * 2;
    v2f a, bm;
    a.x = At[lrow][kk];
    a.y = At[lrow][kk + 1];
    const float* bp = wb + (size_t)kk * DD + ncol;
    bm.x = bp[0];
    bm.y = bp[DD];
    acc = wmma_f32(a, bm, acc);
  }
  float* outp = isQ ? Q : P;
  #pragma unroll
  for (int v = 0; v < 8; ++v) {
    int m = v + halfsel * 8;
    outp[(size_t)(rt * 16 + m) * DD + ncol] = acc[v];
  }
}

// out[r] = leaky( leaky(h[r]@W1 + b1) @ W2 + b2 ), one block (128 thr) per row.
__global__ __launch_bounds__(128) void head_kernel(
    const float* __restrict__ h, const float* __restrict__ w1,
    const float* __restrict__ b1, const float* __restrict__ w2,
    const float* __restrict__ b2, float* __restrict__ out) {
  __shared__ float red[128];
  const int r = blockIdx.x;
  const int t = threadIdx.x;
  const float* hr = h + (size_t)r * DD;
  float acc = b1[t];
  #pragma unroll 8
  for (int k = 0; k < DD; ++k) acc += hr[k] * w1[(size_t)k * 128 + t];
  acc = leaky(acc);
  red[t] = acc * w2[t];
  __syncthreads();
  for (int s = 64; s > 0; s >>= 1) {
    if (t < s) red[t] += red[t + s];
    __syncthreads();
  }
  if (t == 0) out[r] = leaky(red[0] + b2[0]);
}

extern "C" void kernel_launch(void* const* d_in, const int* in_sizes, int n_in,
                              void* d_out, int out_size, void* d_ws, size_t ws_size,
                              hipStream_t stream) {
  (void)in_sizes; (void)n_in; (void)out_size; (void)ws_size;

  const float* feature = (const float*)d_in[0];
  const float* valid   = (const float*)d_in[1];
  const float* l1_w    = (const float*)d_in[2];
  const float* l1_b    = (const float*)d_in[3];
  const float* out1_w  = (const float*)d_in[4];
  const float* out1_b  = (const float*)d_in[5];
  const float* out2_w  = (const float*)d_in[6];
  const float* out2_b  = (const float*)d_in[7];
  const float* lw[4] = {(const float*)d_in[8],  (const float*)d_in[10],
                        (const float*)d_in[12], (const float*)d_in[14]};
  const float* lb[4] = {(const float*)d_in[9],  (const float*)d_in[11],
                        (const float*)d_in[13], (const float*)d_in[15]};
  const float* mlw[5] = {(const float*)d_in[16], (const float*)d_in[18],
                         (const float*)d_in[20], (const float*)d_in[22],
                         (const float*)d_in[24]};
  const float* mlb[5] = {(const float*)d_in[17], (const float*)d_in[19],
                         (const float*)d_in[21], (const float*)d_in[23],
                         (const float*)d_in[25]};

  float* ws = (float*)d_ws;
  float* h = ws;                              // [1024,256] = 1 MB
  float* P = ws + (size_t)BB * NN * DD;       // 1 MB
  float* Q = P + (size_t)BB * NN * DD;        // 1 MB

  // Layer 1: fused (feature @ l1_w) GEMM + GAT
  gat_kernel<true><<<BB * NN, 256, 0, stream>>>(nullptr, nullptr, feature, l1_w,
                                                l1_b, valid, mlw[0], mlb[0], h);
  // Layers 2..5: factorized pairwise_cat GEMM (P/Q) + fused GAT
  for (int l = 0; l < 4; ++l) {
    pq_kernel<<<256, 256, 0, stream>>>(h, lw[l], P, Q);
    gat_kernel<false><<<BB * NN, 256, 0, stream>>>(P, Q, nullptr, nullptr,
                                                   lb[l], valid, mlw[l + 1],
                                                   mlb[l + 1], h);
  }
  // Output MLP head -> [B,N]
  head_kernel<<<BB * NN, 128, 0, stream>>>(h, out1_w, out1_b, out2_w, out2_b,
                                           (float*)d_out);
}